// GaussianMembership_1829656068580
// MI455X (gfx1250) — compile-verified
//
#include <hip/hip_runtime.h>

// Gaussian membership via GEMM decomposition:
//   exponent[n,k] = [x | x^2][n,:] @ W[:,k] + bias[k];  out = exp(exponent)
// W[j,k]   = c[k,j]*a[k,j]        (j < 256)   coefficient of x
// W[256+j] = -0.5*a[k,j]                      coefficient of x^2
// bias[k]  = -0.5 * sum_j c^2 * a ,  a = 1/max(exp(lw),1e-4)^2
//
// GEMM M=32768, Kdim=512, N=64 on V_WMMA_F32_16X16X4_F32 (full f32 precision).
// x tile staged into LDS with GLOBAL_LOAD_ASYNC_TO_LDS_B128 (ASYNCcnt-tracked).

#define N_PTS   32768
#define D_DIM   256
#define K_CL    64
#define KEXT    (2*D_DIM)        // 512
#define ROWS_WG 128              // rows per workgroup (8 waves * 16)
#define THREADS 256
#define XS_STRIDE 260            // padded f32 row stride: 260 % 64 == 4 -> conflict-free

typedef float v2f __attribute__((ext_vector_type(2)));
typedef float v4f __attribute__((ext_vector_type(4)));
typedef float v8f __attribute__((ext_vector_type(8)));
// builtin expects GCC-style vector-of-4-int pointers (per hipcc diagnostic)
typedef int v4i_b __attribute__((vector_size(16)));
typedef __attribute__((address_space(1))) v4i_b* gptr_v4i;
typedef __attribute__((address_space(3))) v4i_b* lptr_v4i;

#if defined(__has_builtin)
#if __has_builtin(__builtin_amdgcn_global_load_async_to_lds_b128) && \
    __has_builtin(__builtin_amdgcn_s_wait_asynccnt)
#define HAVE_ASYNC_LDS 1
#endif
#endif
#ifndef HAVE_ASYNC_LDS
#define HAVE_ASYNC_LDS 0
#endif

// ---------------- prep: build packed W and bias in workspace ----------------
// Wpk layout: [KEXT/2][K_CL][2] floats so a lane's B pair (K=j, j+1) is one b64.
__global__ __launch_bounds__(D_DIM)
void gm_prep(const float* __restrict__ centers, const float* __restrict__ logw,
             float* __restrict__ Wpk, float* __restrict__ bias)
{
    __shared__ float red[D_DIM];
    const int k = blockIdx.x;
    const int d = threadIdx.x;
    const float c  = centers[k * D_DIM + d];
    const float lw = logw[k * D_DIM + d];
    const float w  = fmaxf(__expf(lw), 1e-4f);
    const float a  = 1.0f / (w * w);

    const int s = d & 1;
    Wpk[(((d) >> 1)          * K_CL + k) * 2 + s] = c * a;       // x coeff
    Wpk[(((D_DIM + d) >> 1)  * K_CL + k) * 2 + s] = -0.5f * a;   // x^2 coeff

    red[d] = c * c * a;
    __syncthreads();
    for (int off = D_DIM / 2; off > 0; off >>= 1) {
        if (d < off) red[d] += red[d + off];
        __syncthreads();
    }
    if (d == 0) bias[k] = -0.5f * red[0];
}

// ---------------- main: GEMM + bias + exp ----------------
__global__ __launch_bounds__(THREADS)
void gm_main(const float* __restrict__ x, const float* __restrict__ Wpk,
             const float* __restrict__ bias, float* __restrict__ out)
{
    extern __shared__ float xs[];            // [ROWS_WG][XS_STRIDE]
    const int tid  = threadIdx.x;
    const int lane = tid & 31;
    const int wave = tid >> 5;
    const int rowbase = blockIdx.x * ROWS_WG;

    // ---- stage 128x256 f32 x-tile into LDS ----
    const float* xsrc = x + (size_t)rowbase * D_DIM;
#if HAVE_ASYNC_LDS
    // CDNA5 async direct-to-LDS: no VGPR round-trip, tracked by ASYNCcnt.
    #pragma unroll 8
    for (int c = 0; c < 32; ++c) {
        const int f   = (c * THREADS + tid) * 4;   // flat float index in tile
        const int r   = f >> 8;                    // / D_DIM
        const int col = f & (D_DIM - 1);
        __builtin_amdgcn_global_load_async_to_lds_b128(
            (gptr_v4i)(xsrc + f),
            (lptr_v4i)(&xs[r * XS_STRIDE + col]),
            0, 0);
    }
    __builtin_amdgcn_s_wait_asynccnt(0);
#else
    #pragma unroll 8
    for (int c = 0; c < 32; ++c) {
        const int f   = (c * THREADS + tid) * 4;
        const int r   = f >> 8;
        const int col = f & (D_DIM - 1);
        v4f v = *(const v4f*)(xsrc + f);
        *(v4f*)(&xs[r * XS_STRIDE + col]) = v;
    }
#endif
    __syncthreads();

    // A operand: lane<16 holds row (lane), K pair (j, j+1); lane>=16 row (lane-16), pair (j+2, j+3)
    const int arow = wave * 16 + (lane & 15);
    const int aoff = (lane >> 4) * 2;              // 0 or 2
    const float* arow_ptr = &xs[arow * XS_STRIDE];
    // B operand: col = lane&15 (+16g per group), same K-pair selection
    const int bcol = lane & 15;
    const int bsel = (lane >> 4) * 2;

    v8f acc0 = {}, acc1 = {}, acc2 = {}, acc3 = {};

    // phase 1: j in [0,256): x * (c*a)
    #pragma unroll 4
    for (int j = 0; j < D_DIM; j += 4) {
        v2f a = *(const v2f*)(arow_ptr + j + aoff);
        const float* bp = Wpk + ((size_t)((j + bsel) >> 1) * K_CL + bcol) * 2;
        v2f b0 = *(const v2f*)(bp);
        v2f b1 = *(const v2f*)(bp + 32);
        v2f b2 = *(const v2f*)(bp + 64);
        v2f b3 = *(const v2f*)(bp + 96);
        acc0 = __builtin_amdgcn_wmma_f32_16x16x4_f32(false, a, false, b0, (short)0, acc0, false, false);
        acc1 = __builtin_amdgcn_wmma_f32_16x16x4_f32(false, a, false, b1, (short)0, acc1, false, false);
        acc2 = __builtin_amdgcn_wmma_f32_16x16x4_f32(false, a, false, b2, (short)0, acc2, false, false);
        acc3 = __builtin_amdgcn_wmma_f32_16x16x4_f32(false, a, false, b3, (short)0, acc3, false, false);
    }
    // phase 2: j in [256,512): x^2 * (-0.5*a) -- reuse LDS tile, square on the fly
    #pragma unroll 4
    for (int j = D_DIM; j < KEXT; j += 4) {
        v2f a = *(const v2f*)(arow_ptr + (j - D_DIM) + aoff);
        a = a * a;
        const float* bp = Wpk + ((size_t)((j + bsel) >> 1) * K_CL + bcol) * 2;
        v2f b0 = *(const v2f*)(bp);
        v2f b1 = *(const v2f*)(bp + 32);
        v2f b2 = *(const v2f*)(bp + 64);
        v2f b3 = *(const v2f*)(bp + 96);
        acc0 = __builtin_amdgcn_wmma_f32_16x16x4_f32(false, a, false, b0, (short)0, acc0, false, false);
        acc1 = __builtin_amdgcn_wmma_f32_16x16x4_f32(false, a, false, b1, (short)0, acc1, false, false);
        acc2 = __builtin_amdgcn_wmma_f32_16x16x4_f32(false, a, false, b2, (short)0, acc2, false, false);
        acc3 = __builtin_amdgcn_wmma_f32_16x16x4_f32(false, a, false, b3, (short)0, acc3, false, false);
    }

    // ---- epilogue: bias + exp + store (C/D layout: VGPR v -> row v (+8 upper half), lane -> col) ----
    const float bs0 = bias[bcol];
    const float bs1 = bias[16 + bcol];
    const float bs2 = bias[32 + bcol];
    const float bs3 = bias[48 + bcol];
    const int orow0 = rowbase + wave * 16 + ((lane >> 4) << 3);
    float* obase = out + (size_t)orow0 * K_CL;
    #pragma unroll
    for (int v = 0; v < 8; ++v) {
        float* orow = obase + (size_t)v * K_CL;
        orow[bcol]      = __expf(acc0[v] + bs0);
        orow[16 + bcol] = __expf(acc1[v] + bs1);
        orow[32 + bcol] = __expf(acc2[v] + bs2);
        orow[48 + bcol] = __expf(acc3[v] + bs3);
    }
}

extern "C" void kernel_launch(void* const* d_in, const int* in_sizes, int n_in,
                              void* d_out, int out_size, void* d_ws, size_t ws_size,
                              hipStream_t stream) {
    const float* x       = (const float*)d_in[0];   // [N, D]
    const float* centers = (const float*)d_in[1];   // [K, D]
    const float* logw    = (const float*)d_in[2];   // [K, D]
    float* out  = (float*)d_out;                    // [N, K]
    // workspace: W packed (512*64 f32 = 128KB) + bias (64 f32)
    float* Wpk  = (float*)d_ws;
    float* bias = Wpk + (size_t)KEXT * K_CL;

    gm_prep<<<K_CL, D_DIM, 0, stream>>>(centers, logw, Wpk, bias);

    const size_t smem = (size_t)ROWS_WG * XS_STRIDE * sizeof(float);  // 133,120 B (<320KB WGP LDS)
    (void)hipFuncSetAttribute((const void*)gm_main,
                              hipFuncAttributeMaxDynamicSharedMemorySize, (int)smem);
    gm_main<<<N_PTS / ROWS_WG, THREADS, smem, stream>>>(x, Wpk, bias, out);
}